// SlotAttention_16621523435908
// MI455X (gfx1250) — compile-verified
//
#include <hip/hip_runtime.h>

// ============================================================================
// Slot-attention forward, algebraically reduced for MI455X (gfx1250, wave32).
//
// Key reduction: softmax over a singleton axis == 1.0 exactly, so attn is the
// constant (1+EPS). Hence k/q/dots are dead, upd[b,s,:] = mean_v over batch b
// (same for every slot and every iteration), and since v is affine in LN(x):
//      sum_b v = (sum_b LN(x)) @ w_v + cnt_b * b_v.
// The only O(N) work is one LN+segment-sum pass over x (64 MB -> ~2.8us at
// 23.3 TB/s). All GEMMs run on <=1.5MB state in L2 using fp32 WMMA 16x16x4.
// The GRU GEMM stages its shared 16-row A panel into LDS with the Tensor Data
// Mover (TENSOR_LOAD_TO_LDS + s_wait_tensorcnt), using TDM row padding
// (+2 DWORDs per 64/256) so A-fragment ds_load_b64s are bank-conflict-free.
// ============================================================================

typedef float v2f __attribute__((ext_vector_type(2)));
typedef float v8f __attribute__((ext_vector_type(8)));
typedef unsigned int v4u __attribute__((ext_vector_type(4)));
typedef int v4i __attribute__((ext_vector_type(4)));
typedef int v8i __attribute__((ext_vector_type(8)));

#if defined(__AMDGCN__) && __has_builtin(__builtin_amdgcn_tensor_load_to_lds)
#define USE_TDM 1
#else
#define USE_TDM 0
#endif

static __device__ __forceinline__ v8f wmma_f32_16x16x4(v2f a, v2f b, v8f c) {
  // 8 args: (neg_a, A, neg_b, B, c_mod, C, reuse_a, reuse_b)
  return __builtin_amdgcn_wmma_f32_16x16x4_f32(false, a, false, b, (short)0, c,
                                               false, false);
}

constexpr int N_NODES = 131072;
constexpr int NB      = 256;
constexpr int DIN     = 128;
constexpr int DOUT    = 64;
constexpr int NS      = 8;
constexpr int Hh      = NS * DOUT;   // 512
constexpr int G3      = 3 * Hh;      // 1536
constexpr float EPS_A = 1e-8f;

// ---------------------------------------------------------------------------
// Kernel 1: slots = mu + exp(logsigma) * noise     (d_out is the h buffer)
// ---------------------------------------------------------------------------
__global__ __launch_bounds__(256) void k_init_slots(
    const float* __restrict__ noise, const float* __restrict__ mu,
    const float* __restrict__ lsig, float* __restrict__ h)
{
  int idx = blockIdx.x * 256 + threadIdx.x;
  if (idx < NB * Hh) {
    int d = idx & (DOUT - 1);
    h[idx] = mu[d] + expf(lsig[d]) * noise[idx];
  }
}

// ---------------------------------------------------------------------------
// Kernel 2: per-batch segment sum of LN(x).  One block per batch (batch ids
// are sorted -> binary search the segment).  Deterministic: per-wave register
// accumulators, fixed-order cross-wave reduce, no atomics.
// ---------------------------------------------------------------------------
static __device__ __forceinline__ int lower_bound_i32(
    const int* __restrict__ a, int n, int key)
{
  int lo = 0, hi = n;
  while (lo < hi) { int mid = (lo + hi) >> 1; if (a[mid] < key) lo = mid + 1; else hi = mid; }
  return lo;
}

__global__ __launch_bounds__(1024) void k_ln_segsum(
    const float* __restrict__ x, const int* __restrict__ batch,
    const float* __restrict__ g, const float* __restrict__ b,
    float* __restrict__ xsum, int* __restrict__ cnt)
{
  __shared__ float colacc[32 * DIN];           // 16 KB: one 128-col acc / wave
  const int tid  = threadIdx.x;
  const int w    = tid >> 5;
  const int lane = tid & 31;
  const int seg  = blockIdx.x;                 // batch id

  const int lo = lower_bound_i32(batch, N_NODES, seg);
  const int hi = lower_bound_i32(batch, N_NODES, seg + 1);

  const float4 gv = *(const float4*)(g + lane * 4);
  const float4 bv = *(const float4*)(b + lane * 4);

  float a0 = 0.f, a1 = 0.f, a2 = 0.f, a3 = 0.f;
  for (int r = lo + w; r < hi; r += 32) {
    const float* xr = x + (size_t)r * DIN;
    if (r + 256 < hi)
      __builtin_prefetch(x + (size_t)(r + 256) * DIN + lane * 4, 0, 1);
    float4 xv = *(const float4*)(xr + lane * 4);       // wave reads 512B row
    float s  = xv.x + xv.y + xv.z + xv.w;
    float s2 = xv.x * xv.x + xv.y * xv.y + xv.z * xv.z + xv.w * xv.w;
    #pragma unroll
    for (int m = 16; m; m >>= 1) { s += __shfl_xor(s, m); s2 += __shfl_xor(s2, m); }
    float mean = s * (1.0f / DIN);
    float var  = s2 * (1.0f / DIN) - mean * mean;      // biased var (jnp.var)
    float rstd = 1.0f / sqrtf(var + 1e-5f);
    a0 += (xv.x - mean) * rstd * gv.x + bv.x;
    a1 += (xv.y - mean) * rstd * gv.y + bv.y;
    a2 += (xv.z - mean) * rstd * gv.z + bv.z;
    a3 += (xv.w - mean) * rstd * gv.w + bv.w;
  }
  colacc[w * DIN + lane * 4 + 0] = a0;
  colacc[w * DIN + lane * 4 + 1] = a1;
  colacc[w * DIN + lane * 4 + 2] = a2;
  colacc[w * DIN + lane * 4 + 3] = a3;
  __syncthreads();

  if (tid < DIN) {
    float t = 0.f;
    #pragma unroll
    for (int ww = 0; ww < 32; ++ww) t += colacc[ww * DIN + tid];
    xsum[seg * DIN + tid] = t;
  }
  if (tid == 0) cnt[seg] = hi - lo;
}

// ---------------------------------------------------------------------------
// Kernel 3: upd[b][64] = Wn_b * (xsum[b] @ w_v) + (cnt_b*Wn_b) * b_v
// WMMA f32 16x16x4.  64 waves = 16 M-tiles x 4 N-tiles, K=128 (32 steps).
// ---------------------------------------------------------------------------
__global__ __launch_bounds__(256) void k_upd(
    const float* __restrict__ xsum, const int* __restrict__ cnt,
    const float* __restrict__ wv, const float* __restrict__ bv,
    float* __restrict__ upd)
{
  const int wave = (blockIdx.x * 256 + threadIdx.x) >> 5;   // 0..63
  const int lane = threadIdx.x & 31;
  const int l15 = lane & 15, hi2 = lane >> 4;
  const int mt = wave >> 2, nt = wave & 3;
  const int arow = mt * 16 + l15;
  const int col  = nt * 16 + l15;

  v8f acc = {};
  for (int k = 0; k < 32; ++k) {
    int kb = 4 * k + 2 * hi2;
    v2f a = *(const v2f*)(xsum + arow * DIN + kb);
    v2f bf = { wv[kb * DOUT + col], wv[(kb + 1) * DOUT + col] };
    acc = wmma_f32_16x16x4(a, bf, acc);
  }
  const float attn = 1.0f + EPS_A;
  const float bvc  = bv[col];
  #pragma unroll
  for (int i = 0; i < 8; ++i) {
    int r = mt * 16 + i + 8 * hi2;
    int c = cnt[r];
    float o = 0.0f;
    if (c > 0) {
      float Wn = attn / ((float)c * attn);   // matches reference Wn
      o = acc[i] * Wn + ((float)c * Wn) * bvc;
    }
    upd[r * DOUT + col] = o;
  }
}

// ---------------------------------------------------------------------------
// Kernel 4: out[256x1536] = A[256x512] @ W^T[512x1536] + bias
//   gi path: A[b][j] = upd[b][j & 63]  (slot-replicated u)  -> amask = 63
//   gh path: A = h                                           -> amask = 511
// 1536 waves = 16 x 96 tiles, K=512 (128 WMMA steps / wave).  All 8 waves of
// a block share one M-tile, so the 16-row A panel is DMA'd once per block
// into LDS by the Tensor Data Mover with row padding (stride 66 / 516 floats)
// so the A-fragment ds_load_b64 is bank-conflict-free.
// ---------------------------------------------------------------------------
__global__ __launch_bounds__(256) void k_gru_gemm(
    const float* __restrict__ A, int astride, int amask,
    int lds_stride, int pad_code, int ash,
    const float* __restrict__ W, const float* __restrict__ bias,
    float* __restrict__ out)
{
  __shared__ float apanel[16 * 516];                    // 33 KB (max: gh panel)
  const int wave = (blockIdx.x * 256 + threadIdx.x) >> 5;   // 0..1535
  const int lane = threadIdx.x & 31;
  const int l15 = lane & 15, hi2 = lane >> 4;
  const int mt = wave / 96, nt = wave % 96;
  const int col = nt * 16 + l15;

#if USE_TDM
  (void)ash;
  if (threadIdx.x < 32) {            // wave 0 issues the DMA (EXEC ignored)
    unsigned long long ga =
        (unsigned long long)(uintptr_t)(A + (size_t)(mt * 16) * astride);
    unsigned int lds = (unsigned int)(uintptr_t)apanel;  // flat[31:0] == LDS addr
    // ---- D# group 0: count=1, lds_addr, global_addr, type=2 ("image")
    v4u g0 = { 1u, lds, (unsigned int)ga,
               (unsigned int)((ga >> 32) & 0x1FFFFFFull) | (2u << 30) };
    // ---- D# group 1: data_size=4B, pad every 2^(pad_code+1) DW by 2 DW,
    //      2-D tensor [256 x astride], tile [16 x astride], stride astride.
    v8i g1 = {
      (int)((2u << 16) | (1u << 20) | ((unsigned)pad_code << 22) | (1u << 25)),
      (int)(((unsigned)astride & 0xffffu) << 16),        // tensor_dim0[15:0]
      (int)((((unsigned)astride >> 16) & 0xffffu) | (256u << 16)), // d0 hi | d1 lo
      (int)(((unsigned)astride & 0xffffu) << 16),        // d1 hi=0 | tile_dim0
      16,                                                // tile_dim1 = 16 rows
      (int)astride,                                      // dim0_stride[31:0]
      0, 0                                               // dim0_stride hi, d1_stride
    };
    v4i gz = { 0, 0, 0, 0 };
#if __clang_major__ >= 23
    v8i gz8 = { 0, 0, 0, 0, 0, 0, 0, 0 };
    __builtin_amdgcn_tensor_load_to_lds(g0, g1, gz, gz, gz8, 0);
#else
    __builtin_amdgcn_tensor_load_to_lds(g0, g1, gz, gz, 0);
#endif
    __builtin_amdgcn_s_wait_tensorcnt((short)0);
  }
  __syncthreads();
#else
  (void)pad_code;
  for (int idx = threadIdx.x; idx < (16 << ash); idx += 256) {
    int r = idx >> ash, c = idx & (astride - 1);
    apanel[r * lds_stride + c + ((c >> 8) << 1)] =
        A[(size_t)(mt * 16 + r) * astride + c];
  }
  __syncthreads();
#endif

  v8f acc = {};
  for (int k = 0; k < 128; ++k) {
    int kb = 4 * k + 2 * hi2;                       // even -> b64 aligned
    int cc = kb & amask;
    v2f a  = *(const v2f*)(&apanel[l15 * lds_stride + cc + ((cc >> 8) << 1)]);
    v2f bf = *(const v2f*)(W + (size_t)col * Hh + kb);   // W[o][kk]
    acc = wmma_f32_16x16x4(a, bf, acc);
  }
  const float bc = bias[col];
  #pragma unroll
  for (int i = 0; i < 8; ++i) {
    int r = mt * 16 + i + 8 * hi2;
    out[r * G3 + col] = acc[i] + bc;
  }
}

// ---------------------------------------------------------------------------
// Kernel 5: GRU gates, elementwise, h updated in place.
// ---------------------------------------------------------------------------
__global__ __launch_bounds__(256) void k_gate(
    const float* __restrict__ gi, const float* __restrict__ gh,
    float* __restrict__ h)
{
  int idx = blockIdx.x * 256 + threadIdx.x;     // < 256*512
  int b = idx >> 9, j = idx & 511;
  const float* gib = gi + b * G3;
  const float* ghb = gh + b * G3;
  float ir = gib[j], iz = gib[512 + j], in = gib[1024 + j];
  float hr = ghb[j], hz = ghb[512 + j], hn = ghb[1024 + j];
  float r = 1.0f / (1.0f + expf(-(ir + hr)));
  float z = 1.0f / (1.0f + expf(-(iz + hz)));
  float n = tanhf(in + r * hn);
  float hv = h[idx];
  h[idx] = (1.0f - z) * n + z * hv;
}

// ---------------------------------------------------------------------------
// Kernel 6: h += relu(LN(h) @ w1 + b1) @ w2 + b2, rows are (b,s) pairs
// (2048 x 64).  One wave per 16-row tile; both GEMMs via fp32 WMMA; the
// stage-1 D->stage-2 A layout change goes through padded LDS.
// ---------------------------------------------------------------------------
__global__ __launch_bounds__(128) void k_mlp(
    float* __restrict__ h,
    const float* __restrict__ w1, const float* __restrict__ b1,
    const float* __restrict__ w2, const float* __restrict__ b2,
    const float* __restrict__ lg, const float* __restrict__ lb)
{
  __shared__ float lnbuf[4][16 * 66];    // 16.5 KB (padded vs bank conflicts)
  __shared__ float hid[4][16 * 130];     // 32.5 KB
  const int w = threadIdx.x >> 5, lane = threadIdx.x & 31;
  const int l15 = lane & 15, hi2 = lane >> 4;
  const int tile = blockIdx.x * 4 + w;   // 0..127
  const int row0 = tile * 16;

  // --- LayerNorm over 64 feats: lane handles row l15, half hi2 (32 elems)
  float xv[32];
  {
    const float* hr = h + (size_t)(row0 + l15) * DOUT + hi2 * 32;
    #pragma unroll
    for (int q = 0; q < 8; ++q) {
      float4 t = *(const float4*)(hr + q * 4);
      xv[q * 4 + 0] = t.x; xv[q * 4 + 1] = t.y;
      xv[q * 4 + 2] = t.z; xv[q * 4 + 3] = t.w;
    }
  }
  float s = 0.f, s2 = 0.f;
  #pragma unroll
  for (int q = 0; q < 32; ++q) { s += xv[q]; s2 += xv[q] * xv[q]; }
  s += __shfl_xor(s, 16); s2 += __shfl_xor(s2, 16);
  float mean = s * (1.0f / 64.0f);
  float var  = s2 * (1.0f / 64.0f) - mean * mean;
  float rstd = 1.0f / sqrtf(var + 1e-5f);
  {
    float* lp = &lnbuf[w][l15 * 66 + hi2 * 32];
    #pragma unroll
    for (int q = 0; q < 32; ++q)
      lp[q] = (xv[q] - mean) * rstd * lg[hi2 * 32 + q] + lb[hi2 * 32 + q];
  }
  __syncthreads();

  // --- stage 1: [16x64] @ w1[64x128]
  v8f acc1[8] = {};
  for (int k = 0; k < 16; ++k) {
    int kb = 4 * k + 2 * hi2;
    v2f a = *(const v2f*)(&lnbuf[w][l15 * 66 + kb]);
    #pragma unroll
    for (int nt = 0; nt < 8; ++nt) {
      int c = nt * 16 + l15;
      v2f bf = { w1[kb * 128 + c], w1[(kb + 1) * 128 + c] };
      acc1[nt] = wmma_f32_16x16x4(a, bf, acc1[nt]);
    }
  }
  #pragma unroll
  for (int nt = 0; nt < 8; ++nt) {
    int c = nt * 16 + l15;
    float bb = b1[c];
    #pragma unroll
    for (int i = 0; i < 8; ++i) {
      float t = acc1[nt][i] + bb;
      hid[w][(i + 8 * hi2) * 130 + c] = t > 0.f ? t : 0.f;   // relu
    }
  }
  __syncthreads();

  // --- stage 2: [16x128] @ w2[128x64] + residual
  v8f acc2[4] = {};
  for (int k = 0; k < 32; ++k) {
    int kb = 4 * k + 2 * hi2;
    v2f a = *(const v2f*)(&hid[w][l15 * 130 + kb]);
    #pragma unroll
    for (int nt = 0; nt < 4; ++nt) {
      int c = nt * 16 + l15;
      v2f bf = { w2[kb * DOUT + c], w2[(kb + 1) * DOUT + c] };
      acc2[nt] = wmma_f32_16x16x4(a, bf, acc2[nt]);
    }
  }
  #pragma unroll
  for (int nt = 0; nt < 4; ++nt) {
    int c = nt * 16 + l15;
    float bb = b2[c];
    #pragma unroll
    for (int i = 0; i < 8; ++i) {
      int r = row0 + i + 8 * hi2;
      h[(size_t)r * DOUT + c] += acc2[nt][i] + bb;
    }
  }
}

// ---------------------------------------------------------------------------
extern "C" void kernel_launch(void* const* d_in, const int* in_sizes, int n_in,
                              void* d_out, int out_size, void* d_ws, size_t ws_size,
                              hipStream_t stream)
{
  (void)in_sizes; (void)n_in; (void)out_size; (void)ws_size;

  const float* x     = (const float*)d_in[0];
  const int*   batch = (const int*)  d_in[1];
  const float* noise = (const float*)d_in[2];
  const float* mu    = (const float*)d_in[3];
  const float* lsig  = (const float*)d_in[4];
  // d_in[5..8] = w_q,b_q,w_k,b_k : dead code (softmax over singleton axis)
  const float* wv    = (const float*)d_in[9];
  const float* bv    = (const float*)d_in[10];
  const float* wih   = (const float*)d_in[11];
  const float* whh   = (const float*)d_in[12];
  const float* bih   = (const float*)d_in[13];
  const float* bhh   = (const float*)d_in[14];
  const float* w1    = (const float*)d_in[15];
  const float* b1    = (const float*)d_in[16];
  const float* w2    = (const float*)d_in[17];
  const float* b2    = (const float*)d_in[18];
  const float* ling  = (const float*)d_in[19];
  const float* linb  = (const float*)d_in[20];
  // d_in[21..22] = ln_s_g/b : dead code
  const float* lfg   = (const float*)d_in[23];
  const float* lfb   = (const float*)d_in[24];

  float* h  = (float*)d_out;                 // [256][512] slots / GRU hidden
  float* ws = (float*)d_ws;
  float* xsum = ws;                          // 256*128   = 32768 f
  float* upd  = ws + 32768;                  // 256*64    = 16384 f
  float* gi   = ws + 49152;                  // 256*1536  = 393216 f
  float* gh   = ws + 442368;                 // 256*1536  = 393216 f
  int*   cnt  = (int*)(ws + 835584);         // 256 i32   (total ~3.2 MB)

  k_init_slots<<<(NB * Hh) / 256, 256, 0, stream>>>(noise, mu, lsig, h);
  k_ln_segsum<<<NB, 1024, 0, stream>>>(x, batch, ling, linb, xsum, cnt);
  k_upd<<<8, 256, 0, stream>>>(xsum, cnt, wv, bv, upd);
  // gi is iteration-invariant (upd is): compute once.  K=512 with col&63 A,
  // TDM panel 16x64, pad 2 DW every 64 DW -> LDS stride 66.
  k_gru_gemm<<<192, 256, 0, stream>>>(upd, DOUT, 63, 66, 5, 6, wih, bih, gi);

  for (int it = 0; it < 3; ++it) {
    // gh: TDM panel 16x512, pad 2 DW every 256 DW -> LDS stride 516.
    k_gru_gemm<<<192, 256, 0, stream>>>(h, Hh, 511, 516, 7, 9, whh, bhh, gh);
    k_gate<<<(NB * Hh) / 256, 256, 0, stream>>>(gi, gh, h);
    k_mlp<<<32, 128, 0, stream>>>(h, w1, b1, w2, b2, lfg, lfb);
  }
}